// Model_88923002896473
// MI455X (gfx1250) — compile-verified
//
#include <hip/hip_runtime.h>

typedef __attribute__((ext_vector_type(2))) float v2f;
typedef __attribute__((ext_vector_type(8))) float v8f;

#define Bn   4096
#define Tn   25
#define Kn   8
#define OBS  64
#define ACT  16
#define HID  512
#define IN1  80      // OBS + ACT
#define OUTD 129     // 2*OBS + 1
#define ROWS 32      // M rows per workgroup
#define MAXBLK 136            // 4096/32 + 8 (worst-case per-ensemble padding)
#define NPAD (MAXBLK * ROWS)  // 4352 padded rows
#define KC   256              // K-chunk staged in LDS for 512-deep layers
#define OSTR 145              // LDS stride for 32x129 output tile

__device__ __forceinline__ float swish_f(float x) {
    return x * __builtin_amdgcn_rcpf(1.0f + __expf(-x));
}
__device__ __forceinline__ float softplus_f(float x) {
    return fmaxf(x, 0.0f) + log1pf(__expf(-fabsf(x)));
}
__device__ __forceinline__ v8f wmma4(v8f acc, v2f a, v2f b) {
    return __builtin_amdgcn_wmma_f32_16x16x4_f32(
        false, a, false, b, (short)0, acc, false, false);
}
// async global -> LDS copy of 16 bytes (per lane), CDNA5 ASYNCcnt path
__device__ __forceinline__ void async_g2l_b128(void* lds_ptr, const void* gptr) {
    unsigned lds_off = (unsigned)(size_t)lds_ptr;              // low 32 = LDS offset
    unsigned long long ga = (unsigned long long)(size_t)gptr;
    asm volatile("global_load_async_to_lds_b128 %0, %1, off"
                 :: "v"(lds_off), "v"(ga) : "memory");
}
__device__ __forceinline__ void wait_async0() {
    asm volatile("s_wait_asynccnt 0" ::: "memory");
}

// ---------------------------------------------------------------------------
// One-time: partition batch rows by ensemble into 32-row blocks.
// ---------------------------------------------------------------------------
__global__ void setup_order(const int* __restrict__ assign,
                            int* __restrict__ ord, int* __restrict__ blkK) {
    __shared__ int cnt[Kn], base[Kn], pos[Kn];
    const int tid = threadIdx.x;
    if (tid < Kn) { cnt[tid] = 0; pos[tid] = 0; }
    __syncthreads();
    for (int i = tid; i < NPAD; i += blockDim.x)   ord[i]  = -1;
    for (int i = tid; i < MAXBLK; i += blockDim.x) blkK[i] = -1;
    for (int b = tid; b < Bn; b += blockDim.x)
        atomicAdd(&cnt[assign[b] & 7], 1);
    __syncthreads();
    if (tid == 0) {
        int run = 0, t = 0;
        for (int k = 0; k < Kn; ++k) {
            base[k] = run;
            int nb = (cnt[k] + ROWS - 1) / ROWS;
            for (int j = 0; j < nb; ++j) blkK[t++] = k;
            run += nb * ROWS;
        }
    }
    __syncthreads();
    for (int b = tid; b < Bn; b += blockDim.x) {
        int k = assign[b] & 7;
        ord[base[k] + atomicAdd(&pos[k], 1)] = b;
    }
}

__global__ void init_obs(const float* __restrict__ obs, float* __restrict__ curObs) {
    int i = blockIdx.x * blockDim.x + threadIdx.x;
    if (i < Bn * OBS) curObs[i] = obs[i];
}

// ---------------------------------------------------------------------------
// One-time weight repack: Wr[k][q][n][kk] = W[k][4q+kk][n]  (kk = 0..3).
// Makes each lane's B fragment (two consecutive K rows) a contiguous b64,
// and a full wave's fragment load one contiguous 256B span.
// ---------------------------------------------------------------------------
template <int INDIM, int OUTDIM>
__global__ void repack_w(const float* __restrict__ W, float* __restrict__ Wr) {
    size_t i = (size_t)blockIdx.x * blockDim.x + threadIdx.x;
    if (i >= (size_t)Kn * INDIM * OUTDIM) return;
    int kk = (int)(i & 3);
    size_t r = i >> 2;
    int n = (int)(r % OUTDIM); r /= OUTDIM;
    int q = (int)(r % (INDIM / 4));
    int k = (int)(r / (INDIM / 4));
    Wr[i] = W[((size_t)k * INDIM + 4 * q + kk) * OUTDIM + n];
}

// ---------------------------------------------------------------------------
// Layer 1: h1 = swish( [curObs | act_t] @ W1[k] + b1[k] )   (32x80)x(80x512)
// ---------------------------------------------------------------------------
__global__ void __launch_bounds__(256) gemm_layer1(
    const float* __restrict__ curObs, const float* __restrict__ act,
    const float* __restrict__ W1r, const float* __restrict__ b1,
    const int* __restrict__ ord, const int* __restrict__ blkK,
    float* __restrict__ h1, int t)
{
    constexpr int LSTR = IN1 + 4;  // 84
    __shared__ float Asm[ROWS * LSTR];
    const int blk = blockIdx.x;
    const int kens = blkK[blk];
    if (kens < 0) return;
    const int tid = threadIdx.x;

    for (int i = tid; i < ROWS * IN1; i += 256) {
        int r = i / IN1, c = i - r * IN1;
        int b = ord[blk * ROWS + r];
        float v = 0.0f;
        if (b >= 0)
            v = (c < OBS) ? curObs[b * OBS + c]
                          : act[(b * Tn + t) * ACT + (c - OBS)];
        Asm[r * LSTR + c] = v;
    }
    __syncthreads();

    const int lane = tid & 31, wave = tid >> 5;
    const int m = lane & 15, koff = (lane >> 4) * 2;
    const float* Wk = W1r + (size_t)kens * IN1 * HID;

    v8f acc[4][2];
    #pragma unroll
    for (int j = 0; j < 4; ++j) { acc[j][0] = {}; acc[j][1] = {}; }

    for (int k0 = 0; k0 < IN1; k0 += 4) {
        v2f a0, a1;
        a0.x = Asm[m * LSTR + k0 + koff];
        a0.y = Asm[m * LSTR + k0 + koff + 1];
        a1.x = Asm[(m + 16) * LSTR + k0 + koff];
        a1.y = Asm[(m + 16) * LSTR + k0 + koff + 1];
        const float* wp =
            Wk + (((size_t)(k0 >> 2) * HID + wave * 16 + m) << 2) + koff;
        #pragma unroll
        for (int j = 0; j < 4; ++j) {
            v2f bf = *(const v2f*)(wp + j * 512);   // 128 cols * 4 interleave
            acc[j][0] = wmma4(acc[j][0], a0, bf);
            acc[j][1] = wmma4(acc[j][1], a1, bf);
        }
    }

    const int r0 = blk * ROWS + ((lane >> 4) << 3);
    #pragma unroll
    for (int j = 0; j < 4; ++j) {
        const int n = (wave + 8 * j) * 16 + m;
        const float bv = b1[kens * HID + n];
        #pragma unroll
        for (int v = 0; v < 8; ++v) {
            h1[(size_t)(r0 + v) * HID + n]      = swish_f(acc[j][0][v] + bv);
            h1[(size_t)(r0 + 16 + v) * HID + n] = swish_f(acc[j][1][v] + bv);
        }
    }
}

// ---------------------------------------------------------------------------
// Layer 2: h2 = swish( h1 @ W2[k] + b2[k] )   (32x512)x(512x512)
// K staged in two 256-wide LDS chunks via async global->LDS b128.
// ---------------------------------------------------------------------------
__global__ void __launch_bounds__(256) gemm_layer2(
    const float* __restrict__ h1,
    const float* __restrict__ W2r, const float* __restrict__ b2,
    const int* __restrict__ blkK, float* __restrict__ h2)
{
    constexpr int LSTR = KC + 4;  // 260 floats; row pitch 1040B (16B aligned)
    __shared__ float Asm[ROWS * LSTR];
    const int blk = blockIdx.x;
    const int kens = blkK[blk];
    if (kens < 0) return;
    const int tid = threadIdx.x;
    const int lane = tid & 31, wave = tid >> 5;
    const int m = lane & 15, koff = (lane >> 4) * 2;
    const float* Wk = W2r + (size_t)kens * HID * HID;

    v8f acc[4][2];
    #pragma unroll
    for (int j = 0; j < 4; ++j) { acc[j][0] = {}; acc[j][1] = {}; }

    for (int chunk = 0; chunk < HID / KC; ++chunk) {
        for (int i = tid * 4; i < ROWS * KC; i += 256 * 4) {
            int r = i / KC, c = i - r * KC;
            async_g2l_b128(&Asm[r * LSTR + c],
                           &h1[(size_t)(blk * ROWS + r) * HID + chunk * KC + c]);
        }
        wait_async0();
        __syncthreads();
        for (int k0 = 0; k0 < KC; k0 += 4) {
            v2f a0, a1;
            a0.x = Asm[m * LSTR + k0 + koff];
            a0.y = Asm[m * LSTR + k0 + koff + 1];
            a1.x = Asm[(m + 16) * LSTR + k0 + koff];
            a1.y = Asm[(m + 16) * LSTR + k0 + koff + 1];
            const float* wp =
                Wk + (((size_t)((chunk * KC + k0) >> 2) * HID + wave * 16 + m) << 2)
                   + koff;
            #pragma unroll
            for (int j = 0; j < 4; ++j) {
                v2f bf = *(const v2f*)(wp + j * 512);
                acc[j][0] = wmma4(acc[j][0], a0, bf);
                acc[j][1] = wmma4(acc[j][1], a1, bf);
            }
        }
        __syncthreads();
    }

    const int r0 = blk * ROWS + ((lane >> 4) << 3);
    #pragma unroll
    for (int j = 0; j < 4; ++j) {
        const int n = (wave + 8 * j) * 16 + m;
        const float bv = b2[kens * HID + n];
        #pragma unroll
        for (int v = 0; v < 8; ++v) {
            h2[(size_t)(r0 + v) * HID + n]      = swish_f(acc[j][0][v] + bv);
            h2[(size_t)(r0 + 16 + v) * HID + n] = swish_f(acc[j][1][v] + bv);
        }
    }
}

// ---------------------------------------------------------------------------
// Layer 3 + head, fused:  out = h2 @ W3[k] + b3[k]  (32x512)x(512x129)
// ---------------------------------------------------------------------------
__global__ void __launch_bounds__(256) gemm_layer3_head(
    const float* __restrict__ h2,
    const float* __restrict__ W3r, const float* __restrict__ b3,
    const float* __restrict__ maxlv, const float* __restrict__ minlv,
    const float* __restrict__ noise,
    const int* __restrict__ ord, const int* __restrict__ blkK,
    float* __restrict__ curObs,
    float* __restrict__ obsOut, float* __restrict__ rewOut, int t)
{
    constexpr int LSTR = KC + 4;
    __shared__ float Asm[ROWS * LSTR];     // 33.3 KB
    __shared__ float Osm[ROWS * OSTR];     // 18.6 KB
    const int blk = blockIdx.x;
    const int kens = blkK[blk];
    if (kens < 0) return;
    const int tid = threadIdx.x;
    const int lane = tid & 31, wave = tid >> 5;
    const int m = lane & 15, koff = (lane >> 4) * 2;
    const float* Wk = W3r + (size_t)kens * HID * OUTD;

    v8f acc[2][2];
    #pragma unroll
    for (int j = 0; j < 2; ++j) { acc[j][0] = {}; acc[j][1] = {}; }

    for (int chunk = 0; chunk < HID / KC; ++chunk) {
        for (int i = tid * 4; i < ROWS * KC; i += 256 * 4) {
            int r = i / KC, c = i - r * KC;
            async_g2l_b128(&Asm[r * LSTR + c],
                           &h2[(size_t)(blk * ROWS + r) * HID + chunk * KC + c]);
        }
        wait_async0();
        __syncthreads();
        for (int k0 = 0; k0 < KC; k0 += 4) {
            v2f a0, a1;
            a0.x = Asm[m * LSTR + k0 + koff];
            a0.y = Asm[m * LSTR + k0 + koff + 1];
            a1.x = Asm[(m + 16) * LSTR + k0 + koff];
            a1.y = Asm[(m + 16) * LSTR + k0 + koff + 1];
            const size_t q = (size_t)((chunk * KC + k0) >> 2);
            #pragma unroll
            for (int j = 0; j < 2; ++j) {
                const int ct = wave + 8 * j;
                if (ct < 9) {
                    const int n = ct * 16 + m;
                    v2f bf = {};
                    if (n < OUTD)
                        bf = *(const v2f*)(Wk + ((q * OUTD + n) << 2) + koff);
                    acc[j][0] = wmma4(acc[j][0], a0, bf);
                    acc[j][1] = wmma4(acc[j][1], a1, bf);
                }
            }
        }
        __syncthreads();
    }

    const int r0 = (lane >> 4) << 3;
    #pragma unroll
    for (int j = 0; j < 2; ++j) {
        const int ct = wave + 8 * j;
        if (ct < 9) {
            const int n = ct * 16 + m;
            if (n < OUTD) {
                const float bv = b3[kens * OUTD + n];
                #pragma unroll
                for (int v = 0; v < 8; ++v) {
                    Osm[(r0 + v) * OSTR + n]      = acc[j][0][v] + bv;
                    Osm[(r0 + 16 + v) * OSTR + n] = acc[j][1][v] + bv;
                }
            }
        }
    }
    __syncthreads();

    for (int i = tid; i < ROWS * OBS; i += 256) {
        int pl = i >> 6, d = i & 63;
        int b = ord[blk * ROWS + pl];
        if (b < 0) continue;
        float mean = Osm[pl * OSTR + d];
        float lraw = Osm[pl * OSTR + OBS + d];
        float ml = maxlv[d], mn = minlv[d];
        float lv = ml - softplus_f(ml - lraw);
        lv = mn + softplus_f(lv - mn);
        float nz = noise[(((size_t)t * Kn + kens) * Bn + b) * OBS + d];
        float nx = mean + nz * __expf(0.5f * lv);
        curObs[b * OBS + d] = nx;
        obsOut[((size_t)b * Tn + t) * OBS + d] = nx;
        if (d == 0) rewOut[(size_t)b * Tn + t] = Osm[pl * OSTR + 2 * OBS];
    }
}

// ---------------------------------------------------------------------------
extern "C" void kernel_launch(void* const* d_in, const int* in_sizes, int n_in,
                              void* d_out, int out_size, void* d_ws, size_t ws_size,
                              hipStream_t stream)
{
    const float* obs    = (const float*)d_in[0];
    const float* act    = (const float*)d_in[1];
    const float* W1     = (const float*)d_in[2];
    const float* b1     = (const float*)d_in[3];
    const float* W2     = (const float*)d_in[4];
    const float* b2     = (const float*)d_in[5];
    const float* W3     = (const float*)d_in[6];
    const float* b3     = (const float*)d_in[7];
    const float* maxlv  = (const float*)d_in[8];
    const float* minlv  = (const float*)d_in[9];
    const float* noise  = (const float*)d_in[10];
    const int*   assign = (const int*)d_in[11];

    char* ws = (char*)d_ws;
    int*   ord    = (int*)ws;   ws += ((NPAD * 4 + 255) / 256) * 256;
    int*   blkK   = (int*)ws;   ws += ((MAXBLK * 4 + 255) / 256) * 256;
    float* curObs = (float*)ws; ws += (size_t)Bn * OBS * 4;
    float* h1     = (float*)ws; ws += (size_t)NPAD * HID * 4;
    float* h2     = (float*)ws; ws += (size_t)NPAD * HID * 4;
    float* W1r    = (float*)ws; ws += (size_t)Kn * IN1 * HID * 4;
    float* W2r    = (float*)ws; ws += (size_t)Kn * HID * HID * 4;
    float* W3r    = (float*)ws; ws += (size_t)Kn * HID * OUTD * 4;

    float* obsOut = (float*)d_out;
    float* rewOut = obsOut + (size_t)Bn * Tn * OBS;

    setup_order<<<1, 256, 0, stream>>>(assign, ord, blkK);
    init_obs<<<(Bn * OBS + 255) / 256, 256, 0, stream>>>(obs, curObs);

    const size_t e1 = (size_t)Kn * IN1 * HID;
    const size_t e2 = (size_t)Kn * HID * HID;
    const size_t e3 = (size_t)Kn * HID * OUTD;
    repack_w<IN1, HID><<<(int)((e1 + 255) / 256), 256, 0, stream>>>(W1, W1r);
    repack_w<HID, HID><<<(int)((e2 + 255) / 256), 256, 0, stream>>>(W2, W2r);
    repack_w<HID, OUTD><<<(int)((e3 + 255) / 256), 256, 0, stream>>>(W3, W3r);

    for (int t = 0; t < Tn; ++t) {
        gemm_layer1<<<MAXBLK, 256, 0, stream>>>(
            curObs, act, W1r, b1, ord, blkK, h1, t);
        gemm_layer2<<<MAXBLK, 256, 0, stream>>>(h1, W2r, b2, blkK, h2);
        gemm_layer3_head<<<MAXBLK, 256, 0, stream>>>(
            h2, W3r, b3, maxlv, minlv, noise, ord, blkK,
            curObs, obsOut, rewOut, t);
    }
}